// ReuseLayerNet_1580547971655
// MI455X (gfx1250) — compile-verified
//
#include <hip/hip_runtime.h>

// ---------------------------------------------------------------------------
// ReuseLayerNet fused kernel for MI455X (gfx1250).
//
// The six 1x1 convs collapse into one per-pixel affine transform per tensor:
//   o1 = Ax*x - Ay*y + d,   o2 = -o1
// Setup kernel (1 wave) composes the 3x3 matrices with V_WMMA_F32_16X16X4_F32
// (full f32 precision); main kernel is a single bandwidth-limited streaming
// pass (~403 MB -> ~17 us floor @ 23.3 TB/s) with float4 NT loads/stores.
// ---------------------------------------------------------------------------

typedef __attribute__((ext_vector_type(2))) float v2f;
typedef __attribute__((ext_vector_type(4))) float f4;
typedef __attribute__((ext_vector_type(8))) float v8f;

// Build A-matrix fragment (16x4 f32, top-left 3x3 = S, rest 0).
// Documented layout: lanes 0-15 hold M=0..15; VGPR0 = K=0 (lo lanes) / K=2
// (hi lanes); VGPR1 = K=1 / K=3.
__device__ __forceinline__ v2f afrag3(const float* S, int lane) {
    const int r  = lane & 15;
    const int k0 = (lane < 16) ? 0 : 2;
    v2f a;
    a.x = (r < 3) ? S[r * 3 + k0] : 0.0f;                    // k0 in {0,2} < 3
    a.y = (r < 3 && (k0 + 1) < 3) ? S[r * 3 + k0 + 1] : 0.0f; // only K=1 valid
    return a;
}

// Build B-matrix fragment (4x16 f32, top-left 3x3 = S, rest 0).
// Rows striped across lanes: VGPR0 = K=0 (lanes 0-15) / K=2 (lanes 16-31),
// VGPR1 = K=1 / K=3 (mirrors the documented A/C striping pattern).
__device__ __forceinline__ v2f bfrag3(const float* S, int lane) {
    const int n  = lane & 15;
    const int k0 = (lane < 16) ? 0 : 2;
    v2f b;
    b.x = (n < 3) ? S[k0 * 3 + n] : 0.0f;
    b.y = (n < 3 && (k0 + 1) < 3) ? S[(k0 + 1) * 3 + n] : 0.0f;
    return b;
}

// OUT = M1 * M2 (3x3, row-major) using one V_WMMA_F32_16X16X4_F32.
// C/D layout: VGPR r holds row M=r in lanes 0-15, so element (r,c) for
// r,c < 3 lives in c[r] of lane c.
__device__ __forceinline__ void mat3mul_wmma(float* OUT, const float* M1,
                                             const float* M2, int lane) {
    v2f a = afrag3(M1, lane);
    v2f b = bfrag3(M2, lane);
    v8f c = {};
    c = __builtin_amdgcn_wmma_f32_16x16x4_f32(
        /*neg_a=*/false, a, /*neg_b=*/false, b,
        /*c_mod=*/(short)0, c, /*reuse_a=*/false, /*reuse_b=*/false);
    if (lane < 3) {
        OUT[0 * 3 + lane] = c[0];
        OUT[1 * 3 + lane] = c[1];
        OUT[2 * 3 + lane] = c[2];
    }
    __syncthreads();
}

// One wave (EXEC all-ones, as WMMA requires). Composes all coefficients into
// coef[21]: [0..8]=Ax row-major, [9..17]=-Ay row-major, [18..20]=d.
__global__ void setup_coefs(const float* __restrict__ w1, const float* __restrict__ b1,
                            const float* __restrict__ w2, const float* __restrict__ b2,
                            float* __restrict__ coef) {
    __shared__ float W1[9], W2[9], W1_2[9], W1_3[9], W2_2[9], W2_3[9], AX[9], AY[9];
    const int lane = threadIdx.x;

    if (lane < 9) { W1[lane] = w1[lane]; W2[lane] = w2[lane]; }
    __syncthreads();

    mat3mul_wmma(W1_2, W1, W1, lane);     // W1^2
    mat3mul_wmma(W1_3, W1, W1_2, lane);   // W1^3
    mat3mul_wmma(W2_2, W2, W2, lane);     // W2^2
    mat3mul_wmma(W2_3, W2, W2_2, lane);   // W2^3
    mat3mul_wmma(AX, W2_3, W1_3, lane);   // Ax = W2^3 W1^3
    mat3mul_wmma(AY, W1_3, W2_2, lane);   // Ay = W1^3 W2^2

    if (lane == 0) {
        float c1v[3], c2v[3], ev[3], dx[3], dy[3];
        for (int r = 0; r < 3; ++r) {
            float s1 = 0.0f, s2 = 0.0f, se = 0.0f;
            for (int cc = 0; cc < 3; ++cc) {
                const float id = (r == cc) ? 1.0f : 0.0f;
                s1 += (W1_2[r * 3 + cc] + W1[r * 3 + cc] + id) * b1[cc]; // c1
                s2 += (W2_2[r * 3 + cc] + W2[r * 3 + cc] + id) * b2[cc]; // c2
                se += (W2[r * 3 + cc] + id) * b2[cc];                    // (W2+I)b2
            }
            c1v[r] = s1; c2v[r] = s2; ev[r] = se;
        }
        for (int r = 0; r < 3; ++r) {
            float sx = c2v[r], sy = c1v[r];
            for (int cc = 0; cc < 3; ++cc) {
                sx += W2_3[r * 3 + cc] * c1v[cc];   // dx = W2^3 c1 + c2
                sy += W1_3[r * 3 + cc] * ev[cc];    // dy = W1^3 (W2+I)b2 + c1
            }
            dx[r] = sx; dy[r] = sy;
        }
        for (int i = 0; i < 9; ++i) { coef[i] = AX[i]; coef[9 + i] = -AY[i]; }
        for (int r = 0; r < 3; ++r)  coef[18 + r] = dx[r] - dy[r];
    }
}

// Fused streaming pass. Layout [B=32, C=3, H*W=262144]; each thread owns one
// float4-wide pixel group across all 3 channels of x, y, o1, o2.
__global__ void __launch_bounds__(256) fused_affine(
        const f4* __restrict__ X, const f4* __restrict__ Y,
        const float* __restrict__ coef,
        f4* __restrict__ O1, f4* __restrict__ O2, int n4) {
    constexpr int P4 = (512 * 512) / 4;   // 65536 float4 per channel plane
    const int t = blockIdx.x * blockDim.x + threadIdx.x;
    if (t >= n4) return;

    const int b    = t >> 16;             // t / P4
    const int q    = t & (P4 - 1);
    const int base = b * (3 * P4) + q;

    // Uniform coefficients -> scalar loads, broadcast to the wave.
    float ax[9], ay[9], d0, d1, d2;
#pragma unroll
    for (int i = 0; i < 9; ++i) { ax[i] = coef[i]; ay[i] = coef[9 + i]; }
    d0 = coef[18]; d1 = coef[19]; d2 = coef[20];

    const f4 x0 = __builtin_nontemporal_load(X + base);
    const f4 x1 = __builtin_nontemporal_load(X + base + P4);
    const f4 x2 = __builtin_nontemporal_load(X + base + 2 * P4);
    const f4 y0 = __builtin_nontemporal_load(Y + base);
    const f4 y1 = __builtin_nontemporal_load(Y + base + P4);
    const f4 y2 = __builtin_nontemporal_load(Y + base + 2 * P4);

    const f4 o0 = d0 + ax[0] * x0 + ax[1] * x1 + ax[2] * x2
                     + ay[0] * y0 + ay[1] * y1 + ay[2] * y2;
    const f4 o1 = d1 + ax[3] * x0 + ax[4] * x1 + ax[5] * x2
                     + ay[3] * y0 + ay[4] * y1 + ay[5] * y2;
    const f4 o2 = d2 + ax[6] * x0 + ax[7] * x1 + ax[8] * x2
                     + ay[6] * y0 + ay[7] * y1 + ay[8] * y2;

    __builtin_nontemporal_store(o0, O1 + base);
    __builtin_nontemporal_store(o1, O1 + base + P4);
    __builtin_nontemporal_store(o2, O1 + base + 2 * P4);
    __builtin_nontemporal_store(-o0, O2 + base);
    __builtin_nontemporal_store(-o1, O2 + base + P4);
    __builtin_nontemporal_store(-o2, O2 + base + 2 * P4);
}

extern "C" void kernel_launch(void* const* d_in, const int* in_sizes, int n_in,
                              void* d_out, int out_size, void* d_ws, size_t ws_size,
                              hipStream_t stream) {
    const float* x  = (const float*)d_in[0];
    const float* y  = (const float*)d_in[1];
    const float* w1 = (const float*)d_in[2];
    const float* b1 = (const float*)d_in[3];
    const float* w2 = (const float*)d_in[4];
    const float* b2 = (const float*)d_in[5];
    float* out  = (float*)d_out;
    float* coef = (float*)d_ws;   // 21 floats of composed coefficients

    setup_coefs<<<1, 32, 0, stream>>>(w1, b1, w2, b2, coef);

    const int total = in_sizes[0];        // 32*3*512*512
    const int n4    = total / 12;         // B * (H*W/4) thread tiles
    const int block = 256;
    const int grid  = (n4 + block - 1) / block;   // 8192 blocks

    float* out2 = out + (size_t)out_size / 2;     // second tuple element
    fused_affine<<<grid, block, 0, stream>>>(
        (const f4*)x, (const f4*)y, coef, (f4*)out, (f4*)out2, n4);
}